// QuanvolutionGen_65481071397487
// MI455X (gfx1250) — compile-verified
//
#include <hip/hip_runtime.h>
#include <hip/hip_bf16.h>
#include <math.h>

// Problem constants (from reference)
#define BATCH   4096
#define HW      28
#define NPH     14          // patches per side
#define NP      196         // NPH*NPH
#define NQ      4
#define NL      2
#define KDIM    784         // NP*NQ
#define NCLS    10

typedef __attribute__((ext_vector_type(2))) float v2f;
typedef __attribute__((ext_vector_type(8))) float v8f;

// ---------------------------------------------------------------------------
// Kernel 1: quantum patch simulation -> features[b, p*4+q]  (one thread/patch)
// ---------------------------------------------------------------------------
__global__ void quanv_features_kernel(const int* __restrict__ x,
                                      const float* __restrict__ params,
                                      float* __restrict__ features) {
    int tid = blockIdx.x * blockDim.x + threadIdx.x;
    if (tid >= BATCH * NP) return;

    int b  = tid / NP;
    int p  = tid - b * NP;
    int ph = p / NPH;
    int pw = p - ph * NPH;

    // qubit order: q0=(r0,c0) q1=(r0,c1) q2=(r1,c0) q3=(r1,c1)
    const int base = (b * HW + 2 * ph) * HW + 2 * pw;
    int pix0 = x[base];
    int pix1 = x[base + 1];
    int pix2 = x[base + HW];
    int pix3 = x[base + HW + 1];

    const float K = 3.14159265358979323846f / 510.0f;  // pi/255 * 0.5
    float c0, s0, c1, s1, c2, s2, c3, s3;
    __sincosf((float)pix0 * K, &s0, &c0);
    __sincosf((float)pix1 * K, &s1, &c1);
    __sincosf((float)pix2 * K, &s2, &c2);
    __sincosf((float)pix3 * K, &s3, &c3);

    // state idx = a*8 + b*4 + c*2 + d  (qubit q -> bit (3-q))
    float re[16], im[16];
#pragma unroll
    for (int idx = 0; idx < 16; ++idx) {
        float v = ((idx & 8) ? s0 : c0) * ((idx & 4) ? s1 : c1) *
                  ((idx & 2) ? s2 : c2) * ((idx & 1) ? s3 : c3);
        re[idx] = v;
        im[idx] = 0.0f;
    }

    // CNOT(control mask MC, target mask MT): for bitC=1,bitT=0 swap idx <-> idx|MT
#define CNOT(MC, MT)                                              \
    _Pragma("unroll")                                             \
    for (int idx = 0; idx < 16; ++idx) {                          \
        if ((idx & (MC)) && !(idx & (MT))) {                      \
            int j = idx | (MT);                                   \
            float tr = re[idx]; re[idx] = re[j]; re[j] = tr;      \
            float ti = im[idx]; im[idx] = im[j]; im[j] = ti;      \
        }                                                         \
    }

#pragma unroll
    for (int l = 0; l < NL; ++l) {
        float p0 = params[0 * NL + l];
        float p1 = params[1 * NL + l];
        float p2 = params[2 * NL + l];
        float p3 = params[3 * NL + l];
        // phase: exp(-i*phi/2) on bit=0, exp(+i*phi/2) on bit=1
#pragma unroll
        for (int idx = 0; idx < 16; ++idx) {
            float alpha = 0.5f * (((idx & 8) ? p0 : -p0) +
                                  ((idx & 4) ? p1 : -p1) +
                                  ((idx & 2) ? p2 : -p2) +
                                  ((idx & 1) ? p3 : -p3));
            float sa, ca;
            __sincosf(alpha, &sa, &ca);
            float r = re[idx], i = im[idx];
            re[idx] = r * ca - i * sa;
            im[idx] = r * sa + i * ca;
        }
        CNOT(8, 4)   // q0 -> q1
        CNOT(4, 2)   // q1 -> q2
        CNOT(2, 1)   // q2 -> q3
    }
#undef CNOT

    float z0 = 0.f, z1 = 0.f, z2 = 0.f, z3 = 0.f;
#pragma unroll
    for (int idx = 0; idx < 16; ++idx) {
        float pr = re[idx] * re[idx] + im[idx] * im[idx];
        z0 += (idx & 8) ? -pr : pr;
        z1 += (idx & 4) ? -pr : pr;
        z2 += (idx & 2) ? -pr : pr;
        z3 += (idx & 1) ? -pr : pr;
    }

    float4 out = make_float4(z0, z1, z2, z3);
    *(float4*)(features + (size_t)b * KDIM + p * 4) = out;
}

// ---------------------------------------------------------------------------
// Kernel 2: logits = features @ fc_w^T + fc_b via V_WMMA_F32_16X16X4_F32
// One wave computes a 16(M) x 16(N) tile (N columns 10..15 zero-masked).
// ---------------------------------------------------------------------------
__global__ void wmma_gemm_logits_kernel(const float* __restrict__ features,
                                        const float* __restrict__ fc_w,
                                        const float* __restrict__ fc_b,
                                        float* __restrict__ logits) {
    const int lane = threadIdx.x & 31;
    const int wave = threadIdx.x >> 5;
    const int tile = blockIdx.x * (blockDim.x >> 5) + wave;  // 16-row tile index
    const int half = lane >> 4;        // 0: K=0,1  1: K=2,3
    const int mn   = lane & 15;        // A row / B column within tile

    const int   row   = tile * 16 + mn;                 // A: M index
    const float bmask = (mn < NCLS) ? 1.0f : 0.0f;      // zero invalid classes
    const int   wrow  = (mn < NCLS) ? mn : 0;           // clamped fc_w row

    const float* arow = features + (size_t)row * KDIM + half * 2;
    const float* brow = fc_w + (size_t)wrow * KDIM + half * 2;

    v8f acc = {};
#pragma unroll 4
    for (int k = 0; k < KDIM; k += 4) {
        v2f a, bb;
        a.x  = arow[k];
        a.y  = arow[k + 1];
        bb.x = brow[k] * bmask;
        bb.y = brow[k + 1] * bmask;
        acc = __builtin_amdgcn_wmma_f32_16x16x4_f32(
            /*neg_a=*/false, a, /*neg_b=*/false, bb,
            /*c_mod=*/(short)0, acc, /*reuse_a=*/false, /*reuse_b=*/false);
    }

    // C/D layout: VGPR i, lanes 0-15 -> M=i, N=lane; lanes 16-31 -> M=i+8
    if (mn < NCLS) {
        float bias = fc_b[mn];
#pragma unroll
        for (int i = 0; i < 8; ++i) {
            int m = tile * 16 + half * 8 + i;
            logits[(size_t)m * NCLS + mn] = acc[i] + bias;
        }
    }
}

// ---------------------------------------------------------------------------
// Kernel 3: row-wise log_softmax over 10 classes (one thread per row)
// ---------------------------------------------------------------------------
__global__ void log_softmax_kernel(const float* __restrict__ logits,
                                   float* __restrict__ out) {
    int b = blockIdx.x * blockDim.x + threadIdx.x;
    if (b >= BATCH) return;
    float v[NCLS];
    float mx = -INFINITY;
#pragma unroll
    for (int j = 0; j < NCLS; ++j) {
        v[j] = logits[(size_t)b * NCLS + j];
        mx = fmaxf(mx, v[j]);
    }
    float s = 0.0f;
#pragma unroll
    for (int j = 0; j < NCLS; ++j) s += __expf(v[j] - mx);
    float lse = mx + __logf(s);
#pragma unroll
    for (int j = 0; j < NCLS; ++j) out[(size_t)b * NCLS + j] = v[j] - lse;
}

// ---------------------------------------------------------------------------
extern "C" void kernel_launch(void* const* d_in, const int* in_sizes, int n_in,
                              void* d_out, int out_size, void* d_ws, size_t ws_size,
                              hipStream_t stream) {
    const int*   x      = (const int*)d_in[0];     // (B,1,28,28) int32
    const float* params = (const float*)d_in[1];   // (4,2)
    const float* fc_w   = (const float*)d_in[2];   // (10,784)
    const float* fc_b   = (const float*)d_in[3];   // (10,)
    float* out = (float*)d_out;                    // (B,10)

    float* features = (float*)d_ws;                        // B*784 floats
    float* logits   = features + (size_t)BATCH * KDIM;     // B*10 floats

    // 1) patch simulation -> features
    {
        int total = BATCH * NP;                   // 802816
        int threads = 256;
        int blocks = (total + threads - 1) / threads;
        quanv_features_kernel<<<blocks, threads, 0, stream>>>(x, params, features);
    }
    // 2) WMMA GEMM -> logits. 256 tiles of 16 rows; 8 waves/block of 256 thr.
    {
        int tiles = BATCH / 16;                   // 256
        int threads = 256;                        // 8 waves
        int blocks = tiles / 8;                   // 32
        wmma_gemm_logits_kernel<<<blocks, threads, 0, stream>>>(features, fc_w, fc_b, logits);
    }
    // 3) log_softmax -> out
    {
        int threads = 256;
        int blocks = (BATCH + threads - 1) / threads;
        log_softmax_kernel<<<blocks, threads, 0, stream>>>(logits, out);
    }
}